// MLP_HENN_50010599194898
// MI455X (gfx1250) — compile-verified
//
#include <hip/hip_runtime.h>
#include <math.h>

// ---------------------------------------------------------------------------
// Problem constants (from the reference): IN_DIM=256, HIDDEN=512.
// Phase 1: sorted segment-sum gather  (memory bound, L2-resident x)
// Phase 2: fused  sigmoid(relu(Z@W1+b1)@W2+b2)  using V_WMMA_F32_16X16X4_F32
// ---------------------------------------------------------------------------

static constexpr int IN_DIM = 256;
static constexpr int HIDDEN = 512;

typedef float v2f __attribute__((ext_vector_type(2)));
typedef float v8f __attribute__((ext_vector_type(8)));

// ---------------------------------------------------------------------------
// Kernel A: Z[seg,:] = sum over incidences i with ids[i]==seg of x[nodes[i],:]
// ids are sorted -> binary search this block's range once, broadcast via LDS.
// 64 threads/block, each owns 4 consecutive floats (float4, b128 loads).
// nodes[i] is wave-uniform -> scalar load; row reads are fully coalesced.
// ---------------------------------------------------------------------------
__global__ __launch_bounds__(64) void seg_gather_sum_kernel(
    const float* __restrict__ x, const int* __restrict__ nodes,
    const int* __restrict__ ids, float* __restrict__ Z, int n_inc) {
  const int seg = blockIdx.x;
  __shared__ int s_range[2];
  if (threadIdx.x == 0) {
    int lo = 0, hi = n_inc;                 // lower_bound(ids, seg)
    while (lo < hi) { int mid = (lo + hi) >> 1; if (ids[mid] < seg) lo = mid + 1; else hi = mid; }
    s_range[0] = lo;
    int lo2 = lo, hi2 = n_inc;              // lower_bound(ids, seg+1)
    while (lo2 < hi2) { int mid = (lo2 + hi2) >> 1; if (ids[mid] < seg + 1) lo2 = mid + 1; else hi2 = mid; }
    s_range[1] = lo2;
  }
  __syncthreads();
  const int start = s_range[0], end = s_range[1];

  float4 acc = make_float4(0.f, 0.f, 0.f, 0.f);
  for (int i = start; i < end; ++i) {
    const int node = nodes[i];  // uniform across block -> s_load
    float4 v = ((const float4*)(x + (size_t)node * IN_DIM))[threadIdx.x];
    acc.x += v.x; acc.y += v.y; acc.z += v.z; acc.w += v.w;
  }
  ((float4*)(Z + (size_t)seg * IN_DIM))[threadIdx.x] = acc;
}

// ---------------------------------------------------------------------------
// Kernel B: fused MLP head over Z rows.
//   out[m] = sigmoid( relu(Z[m,:]@W1 + b1) @ W2 + b2 )
// Block: 256 threads = 8 waves, M-tile = 128 rows (16 rows per wave).
// Z tile staged in LDS with +4 float row padding (row stride 260 -> A-fragment
// ds reads hit 16 distinct banks instead of a 16-way conflict).
// Each wave sweeps N in 8 groups of 64 cols (4x 16x16 C tiles live at once,
// 32 accumulator VGPRs), K-loop step 4 with V_WMMA_F32_16X16X4_F32.
// ---------------------------------------------------------------------------
__global__ __launch_bounds__(256) void mlp_head_kernel(
    const float* __restrict__ Z, const float* __restrict__ W1,
    const float* __restrict__ b1, const float* __restrict__ W2,
    const float* __restrict__ b2, float* __restrict__ out, int n_rows) {
  __shared__ float zt[128][IN_DIM + 4];   // ~130 KB of the 320 KB WGP LDS

  const int tid  = threadIdx.x;
  const int wave = tid >> 5;
  const int lane = tid & 31;
  const int half = lane >> 4;             // 0: lanes 0-15, 1: lanes 16-31
  const int lrow = lane & 15;
  const int row0 = blockIdx.x * 128;

  // Cooperative tile load: 128 rows x 256 f32 = 8192 float4, 32 per thread.
  for (int it = 0; it < 32; ++it) {
    const int f  = it * 256 + tid;        // linear float4 index
    const int r  = f >> 6;                // 64 float4 per row
    const int c4 = f & 63;
    float4 v = make_float4(0.f, 0.f, 0.f, 0.f);
    const int gr = row0 + r;
    if (gr < n_rows) v = ((const float4*)(Z + (size_t)gr * IN_DIM))[c4];
    *(float4*)&zt[r][c4 * 4] = v;
  }
  __syncthreads();

  const int wr = wave * 16 + lrow;        // this lane's A-matrix row (M=lane%16)
  float facc[8] = {0.f, 0.f, 0.f, 0.f, 0.f, 0.f, 0.f, 0.f};

  for (int ng = 0; ng < 8; ++ng) {
    const int nbase = ng * 64;
    v8f c[4] = {};                        // 4 C tiles: cols nbase + t*16 + lrow

#if __has_builtin(__builtin_amdgcn_wmma_f32_16x16x4_f32)
    for (int k = 0; k < IN_DIM; k += 4) {
      const int kk = k + 2 * half;        // A layout: half selects K pair
      v2f a;
      a.x = zt[wr][kk];
      a.y = zt[wr][kk + 1];
      // B layout: N = lane%16, VGPR0/1 = rows K=kk, kk+1 of W1
      const float* wp = W1 + (size_t)kk * HIDDEN + nbase + lrow;
      v2f bm[4];
#pragma unroll
      for (int t = 0; t < 4; ++t) {
        bm[t].x = wp[t * 16];
        bm[t].y = wp[t * 16 + HIDDEN];
      }
#pragma unroll
      for (int t = 0; t < 4; ++t) {
        c[t] = __builtin_amdgcn_wmma_f32_16x16x4_f32(
            false, a, false, bm[t], (short)0, c[t], false, false);
      }
    }
#else
    // VALU fallback matching the C-tile lane layout (N=lane%16, M=half*8+v).
    for (int t = 0; t < 4; ++t) {
      const int n = nbase + t * 16 + lrow;
      for (int v = 0; v < 8; ++v) {
        float s = 0.f;
        const int m = wave * 16 + half * 8 + v;
        for (int k = 0; k < IN_DIM; ++k) s += zt[m][k] * W1[(size_t)k * HIDDEN + n];
        c[t][v] = s;
      }
    }
#endif

    // Epilogue for this N-group: +b1, ReLU, fold through W2 column weights.
#pragma unroll
    for (int t = 0; t < 4; ++t) {
      const int n = nbase + t * 16 + lrow;
      const float bb = b1[n];
      const float ww = W2[n];
#pragma unroll
      for (int v = 0; v < 8; ++v) {
        float h = c[t][v] + bb;
        h = h > 0.f ? h : 0.f;
        facc[v] += h * ww;
      }
    }
  }

  // Each lane holds partial sums over 32 of the 512 hidden cols for 8 rows.
  // Reduce across the 16 lanes of each half (disjoint col sets).
#pragma unroll
  for (int v = 0; v < 8; ++v) {
    float s = facc[v];
    s += __shfl_xor(s, 1, 16);
    s += __shfl_xor(s, 2, 16);
    s += __shfl_xor(s, 4, 16);
    s += __shfl_xor(s, 8, 16);
    facc[v] = s;
  }

  if (lrow == 0) {
    const float bb2 = b2[0];
#pragma unroll
    for (int v = 0; v < 8; ++v) {
      const int m = row0 + wave * 16 + half * 8 + v;  // C layout: M = half*8 + v
      if (m < n_rows) out[m] = 1.0f / (1.0f + expf(-(facc[v] + bb2)));
    }
  }
}

// ---------------------------------------------------------------------------
// Inputs (setup_inputs order):
//  0: x (100000*256 f32)   1: target_nodes (1e6 i32)  2: target_ids (1e6 i32)
//  3: num_segments (i32[1], device!)  4: W1 (256*512)  5: b1 (512)
//  6: W2 (512)             7: b2 (1)
// num_segments equals out_size -> use out_size on host (capture-safe).
// ---------------------------------------------------------------------------
extern "C" void kernel_launch(void* const* d_in, const int* in_sizes, int n_in,
                              void* d_out, int out_size, void* d_ws, size_t ws_size,
                              hipStream_t stream) {
  const float* x     = (const float*)d_in[0];
  const int*   nodes = (const int*)d_in[1];
  const int*   ids   = (const int*)d_in[2];
  const float* W1    = (const float*)d_in[4];
  const float* b1    = (const float*)d_in[5];
  const float* W2    = (const float*)d_in[6];
  const float* b2    = (const float*)d_in[7];
  float*       out   = (float*)d_out;

  const int n_inc = in_sizes[1];
  const int n_seg = out_size;           // == num_segments (100000)

  float* Z = (float*)d_ws;              // n_seg * 256 f32 = 102.4 MB scratch
  (void)ws_size; (void)n_in;

  seg_gather_sum_kernel<<<n_seg, 64, 0, stream>>>(x, nodes, ids, Z, n_inc);

  const int mblocks = (n_seg + 127) / 128;
  mlp_head_kernel<<<mblocks, 256, 0, stream>>>(Z, W1, b1, W2, b2, out, n_seg);
}